// RecursiveStack_19559281066378
// MI455X (gfx1250) — compile-verified
//
#include <hip/hip_runtime.h>
#include <cstdint>

// MoE token-choice router for MI455X (gfx1250, wave32).
// GEMM (16384 x 8 x 2048) on the matrix engine via V_WMMA_F32_16X16X4_F32,
// elementwise tail (softmax / entropy / gumbel) on VALU with 8-lane shuffles.
// HBM-bound: 134 MB of x @ 23.3 TB/s ~= 5.8 us floor.
//
// B-fragment padding (experts 8..15) is handled with a zeroed LDS slot and a
// per-lane LDS stride (0 for padding lanes), so the inner loop is pure
// ds_load_2addr_b32 -> v_wmma with no per-iteration cndmask.

typedef float v2f __attribute__((ext_vector_type(2)));
typedef float v8f __attribute__((ext_vector_type(8)));

#define D_DIM 2048
#define E_DIM 8
#define NTOK  16384           // B*S = 4*4096
#define ZPAD  256             // zero slot (floats) for padding-lane B reads
#define TEMP_INV (1.0f / (1.0f + 1e-8f))

__device__ __forceinline__ uint32_t rotl32(uint32_t v, int r) {
    return (v << r) | (v >> (32 - r));
}

// threefry2x32, 16 rounds (JAX's PRNG family), deterministic counter hash.
__device__ __forceinline__ uint2 tf2x32(uint32_t k0, uint32_t k1,
                                        uint32_t c0, uint32_t c1) {
    uint32_t ks2 = 0x1BD11BDAu ^ k0 ^ k1;
    uint32_t x0 = c0 + k0, x1 = c1 + k1;
#define TFR(r) { x0 += x1; x1 = rotl32(x1, r); x1 ^= x0; }
    TFR(13) TFR(15) TFR(26) TFR(6)
    x0 += k1;  x1 += ks2 + 1u;
    TFR(17) TFR(29) TFR(16) TFR(24)
    x0 += ks2; x1 += k0 + 2u;
    TFR(13) TFR(15) TFR(26) TFR(6)
    x0 += k0;  x1 += k1 + 3u;
    TFR(17) TFR(29) TFR(16) TFR(24)
    x0 += k1;  x1 += ks2 + 4u;
#undef TFR
    return make_uint2(x0, x1);
}

__device__ __forceinline__ float bits_to_unit(uint32_t b) {
    return (float)(b >> 8) * (1.0f / 16777216.0f);   // [0, 1)
}

__device__ __forceinline__ float tf_normal(uint32_t seed, uint32_t idx) {
    uint2 r = tf2x32(0u, seed, idx, 0x9E3779B9u);
    float v = 2.0f * bits_to_unit(r.x) - 1.0f;
    v = fminf(fmaxf(v, -0.999999f), 0.999999f);
    return 1.41421356237f * erfinvf(v);
}

__global__ void __launch_bounds__(256)
router_kernel(const float* __restrict__ x, const float* __restrict__ W,
              const float* __restrict__ b, float* __restrict__ out_rw,
              float* __restrict__ out_sp, float* __restrict__ ws_acc) {
    // k-major transposed W in LDS: Wt[k][e], 64 KB, plus 1 KB zero slot.
    __shared__ float Wt[D_DIM * E_DIM + ZPAD];
    const int t = threadIdx.x;
    for (int i = t; i < D_DIM * E_DIM; i += 256) {
        int e = i >> 11;            // i / D_DIM
        int k = i & (D_DIM - 1);
        Wt[k * E_DIM + e] = W[i];
    }
    Wt[D_DIM * E_DIM + t] = 0.0f;   // t in [0,256): fill the zero slot
    __syncthreads();

    const int lane  = t & 31;
    const int wave  = t >> 5;
    const int half  = lane >> 4;    // 0 | 1
    const int e     = lane & 15;    // WMMA N index (expert; 8..15 = padding)
    const int ec    = e & 7;
    const bool evalid = (e < E_DIM);
    const int tok_base = (blockIdx.x * 8 + wave) * 16;

    // A fragment source: row M = lane&15 (token), K pair offset by half*2.
    const float* xrow = x + (size_t)(tok_base + e) * D_DIM + half * 2;
    const float bias = evalid ? b[ec] : 0.0f;

    // B fragment base: valid lanes walk Wt rows; padding lanes park on zeros.
    int woff = evalid ? ((2 * half) * E_DIM + ec) : (D_DIM * E_DIM);
    const int winc = evalid ? (32 * E_DIM) : 0;   // floats per outer iter

    v8f c = {0.f, 0.f, 0.f, 0.f, 0.f, 0.f, 0.f, 0.f};

    for (int k0 = 0; k0 < D_DIM; k0 += 32) {
        __builtin_prefetch(xrow + k0 + 512, 0, 3);   // global_prefetch_b8
#pragma unroll
        for (int kk = 0; kk < 32; kk += 4) {
            // A 16x4 f32 fragment: VGPR0 = K (k+2*half), VGPR1 = K+1.
            v2f a = *reinterpret_cast<const v2f*>(xrow + k0 + kk);
            // B 4x16 f32 fragment: column = lane&15, mirrored K layout.
            v2f bf;
            bf.x = Wt[woff + kk * E_DIM];
            bf.y = Wt[woff + kk * E_DIM + E_DIM];
            c = __builtin_amdgcn_wmma_f32_16x16x4_f32(
                    false, a, false, bf, (short)0, c, false, false);
        }
        woff += winc;
    }

    // C layout: VGPR r, lane l -> logits[tok_base + r + 8*(l>>4)][l&15].
    union { v8f v; float f[8]; } cu;
    cu.v = c;

    float ent_acc = 0.0f, step_acc = 0.0f;
#pragma unroll
    for (int r = 0; r < 8; ++r) {
        const int tok = tok_base + r + 8 * half;
        float logit = cu.f[r] + bias;

        // stabilize: max over the 8 experts (xor 1/2/4 stays in 8-lane group)
        float m = logit;
        m = fmaxf(m, __shfl_xor(m, 1, 32));
        m = fmaxf(m, __shfl_xor(m, 2, 32));
        m = fmaxf(m, __shfl_xor(m, 4, 32));
        logit -= m;

        const uint32_t idx = (uint32_t)tok * E_DIM + (uint32_t)ec;
        logit += 0.05f * tf_normal(1u, idx);
        logit = fminf(fmaxf(logit, -50.0f), 50.0f);

        // soft probabilities
        float z  = logit * TEMP_INV;
        float zm = z;
        zm = fmaxf(zm, __shfl_xor(zm, 1, 32));
        zm = fmaxf(zm, __shfl_xor(zm, 2, 32));
        zm = fmaxf(zm, __shfl_xor(zm, 4, 32));
        float ex = __expf(z - zm);
        float s  = ex;
        s += __shfl_xor(s, 1, 32);
        s += __shfl_xor(s, 2, 32);
        s += __shfl_xor(s, 4, 32);
        float p = ex / s;

        // per-token entropy and expected steps (reduced over experts)
        float ent = -(p * __logf(p + 1e-8f));
        float es  = p * (float)ec;
        float ent_r = ent, es_r = es;
        ent_r += __shfl_xor(ent_r, 1, 32);  es_r += __shfl_xor(es_r, 1, 32);
        ent_r += __shfl_xor(ent_r, 2, 32);  es_r += __shfl_xor(es_r, 2, 32);
        ent_r += __shfl_xor(ent_r, 4, 32);  es_r += __shfl_xor(es_r, 4, 32);
        if (e == 0) { ent_acc += ent_r; step_acc += es_r; }

        // gumbel-softmax routing weights
        uint2 gb = tf2x32(0u, 2u, idx, 0x7F4A7C15u);
        float u2 = bits_to_unit(gb.x);
        float g  = -__logf(-__logf(u2 + 1e-8f) + 1e-8f) * 0.5f;
        float zg = (logit + g) * TEMP_INV;
        float gm = zg;
        gm = fmaxf(gm, __shfl_xor(gm, 1, 32));
        gm = fmaxf(gm, __shfl_xor(gm, 2, 32));
        gm = fmaxf(gm, __shfl_xor(gm, 4, 32));
        float exg = __expf(zg - gm);
        float sg  = exg;
        sg += __shfl_xor(sg, 1, 32);
        sg += __shfl_xor(sg, 2, 32);
        sg += __shfl_xor(sg, 4, 32);
        float rw = exg / sg;

        if (evalid) {
            out_rw[idx] = rw;
            out_sp[idx] = p;
        }
    }

    // fold lane16's accumulator into lane0, one atomic per wave per scalar
    ent_acc  += __shfl_xor(ent_acc, 16, 32);
    step_acc += __shfl_xor(step_acc, 16, 32);
    if (lane == 0) {
        atomicAdd(&ws_acc[0], ent_acc);
        atomicAdd(&ws_acc[1], step_acc);
    }
}

__global__ void zero_ws_kernel(float* ws) {
    if (threadIdx.x < 2) ws[threadIdx.x] = 0.0f;
}

__global__ void finalize_kernel(const float* __restrict__ ws,
                                float* __restrict__ out_scalars) {
    if (threadIdx.x == 0) {
        const float invN = 1.0f / (float)NTOK;
        float ent = ws[0] * invN;
        ent = fminf(fmaxf(ent, 0.0f), 20.0f);
        out_scalars[0] = ent;            // entropy
        out_scalars[1] = ws[1] * invN;   // expected_steps
    }
}

extern "C" void kernel_launch(void* const* d_in, const int* in_sizes, int n_in,
                              void* d_out, int out_size, void* d_ws, size_t ws_size,
                              hipStream_t stream) {
    const float* x = (const float*)d_in[0];   // (B,S,D) f32
    const float* W = (const float*)d_in[1];   // (E,D)   f32
    const float* b = (const float*)d_in[2];   // (E,)    f32

    float* out      = (float*)d_out;
    float* rw       = out;                         // NTOK*E routing_weights
    float* scalars  = out + (size_t)NTOK * E_DIM;  // entropy, expected_steps
    float* sp       = scalars + 2;                 // NTOK*E soft_probs
    float* ws       = (float*)d_ws;

    zero_ws_kernel<<<1, 32, 0, stream>>>(ws);
    router_kernel<<<NTOK / 128, 256, 0, stream>>>(x, W, b, rw, sp, ws);
    finalize_kernel<<<1, 32, 0, stream>>>(ws, scalars);
}